// cabb_68092411511620
// MI455X (gfx1250) — compile-verified
//
#include <hip/hip_runtime.h>
#include <hip/hip_bf16.h>
#include <math.h>

// ---------------------------------------------------------------------------
// Elementwise box solver (no matrix structure -> no WMMA; VALU/TRANS-bound).
// Per (row,axis): 2 fused loops x 12 interleaved 60-step bisection chains.
// Bisection decisions use sign(eps') only, so eps' is evaluated in the
// rcp-free form  sign(t1 + cd/wm) == sign(fma(t1, wm, cd)):  1 TRANS/body.
// ---------------------------------------------------------------------------

#define BISECT_ITERS 60
#define LN2F 0.69314718055994531f

__device__ __forceinline__ float fast_rcp(float x) {
  return __builtin_amdgcn_rcpf(x);            // v_rcp_f32
}

// log(max(w,1e-12)) built from raw v_log_f32 (base 2)
__device__ __forceinline__ float safe_log(float w) {
  return __builtin_amdgcn_logf(fmaxf(w, 1e-12f)) * LN2F;
}

// eps(w, wp, wh) with neg_lwp = -safe_log(wp) hoisted by the caller.
// smooth_l1(x) with beta==1: d<1 ? 0.5 d^2 : d-0.5
__device__ __forceinline__ float eps_fast(float w, float neg_lwp, float wh) {
  float a  = (w - wh) * 0.5f;
  float da = fabsf(a);
  float s1 = (da < 1.0f) ? (0.5f * da * da) : (da - 0.5f);
  float wm = fmaxf(w, 1e-12f);
  float df = fmaf(__builtin_amdgcn_logf(wm), LN2F, neg_lwp); // log w - log wp
  float db = fabsf(df);
  float s2 = (db < 1.0f) ? (0.5f * db * db) : (db - 0.5f);
  return s1 + s2;
}

// Same SIGN as eps'(w) = clip(0.25 w - qwh, -.5, .5) + clip(df,-1,1)/wm,
// multiplied through by wm = max(w,1e-12) > 0 to avoid the v_rcp:
//   sign(eps') == sign( fma(t1, wm, cd) )
__device__ __forceinline__ float eps_prime_sgn(float w, float neg_lwp, float qwh) {
  float t1 = fminf(fmaxf(fmaf(0.25f, w, -qwh), -0.5f), 0.5f); // fma + v_med3
  float wm = fmaxf(w, 1e-12f);
  float df = fmaf(__builtin_amdgcn_logf(wm), LN2F, neg_lwp);  // v_log + fma
  float cd = fminf(fmaxf(df, -1.0f), 1.0f);                   // v_med3
  return fmaf(t1, wm, cd);                                    // fma (no rcp)
}

// Two solve_o1 problems sharing wp, run as 12 interleaved bisection chains.
// Chain layout: [0..5] = problem A, [6..11] = problem B;
// within each: cA, c2, c3 bisect eps' ; c4, c5, c6 bisect sigma = w * eps'.
__device__ void solve_o1_pair(float wp,
                              float whA, float a1A, float b1A,
                              float whB, float a1B, float b1B,
                              float* outA, float* outB) {
  const float E = 2.7182818284590452f;        // exp(beta) = exp(1)
  const float neg_lwp = -safe_log(wp);
  const float ewp = E * wp;

  const float wh[2]  = { whA, whB };
  const float qwh[2] = { 0.25f * whA, 0.25f * whB };
  const float w0[2]  = { b1A - a1A, b1B - a1B };

  float u0[12], v0[12];
#pragma unroll
  for (int p = 0; p < 2; ++p) {
    const float W  = wh[p];
    const float q0 = w0[p];
    const float base  = fmaxf(q0, fmaxf(W - 2.0f, ewp));
    const float base3 = fmaxf(base, 2.0f);    // max(max(w0,2), max(wh-2, e wp))
    const float t     = 1.0f - 32.0f * fast_rcp(fmaxf(W * W, 1e-12f));
    const float disc  = sqrtf(fmaxf(t, 0.0f));
    const float mn    = fminf(ewp, W);
    const int o = p * 6;
    u0[o+0] = fmaxf(q0, W);  v0[o+0] = wp;                               // cA
    u0[o+1] = fmaxf(q0, wp); v0[o+1] = mn;                               // c2
    u0[o+2] = base3;         v0[o+2] = W;                                // c3
    u0[o+3] = base;          v0[o+3] = fminf(E, W);                      // c4
    u0[o+4] = base;          v0[o+4] = fminf(mn, W * 0.25f * (1.0f + disc)); // c5
    u0[o+5] = fmaxf(base, W * 0.25f * (1.0f - disc)); v0[o+5] = mn;      // c6
  }

  // Midpoint-increment bisection: m <- m -/+ h0 * 2^-(k+1). Power-of-two
  // scales are exact, so the interval arithmetic matches (u+v)/2 stepping.
  float m[12], h0[12];
#pragma unroll
  for (int j = 0; j < 12; ++j) {
    m[j]  = 0.5f * (u0[j] + v0[j]);
    h0[j] = 0.5f * (v0[j] - u0[j]);
  }

  float s = 0.5f;
  for (int it = 0; it < BISECT_ITERS; ++it) {
#pragma unroll
    for (int j = 0; j < 12; ++j) {            // 12 independent chains / iter
      const float q = (j < 6) ? qwh[0] : qwh[1];
      float g = eps_prime_sgn(m[j], neg_lwp, q);
      float f = ((j % 6) >= 3) ? m[j] * g : g;     // sigma chains: sign(m*g)
      bool  c = (f >= 0.0f);                       // NaN -> false (jnp.where)
      float sel = c ? -h0[j] : h0[j];
      m[j] = fmaf(sel, s, m[j]);
    }
    s *= 0.5f;
  }

  // Endpoint sign tests (only signs of f(u0), f(v0) are ever used).
  float res[12];
#pragma unroll
  for (int j = 0; j < 12; ++j) {
    const float q = (j < 6) ? qwh[0] : qwh[1];
    float gu = eps_prime_sgn(u0[j], neg_lwp, q);
    float gv = eps_prime_sgn(v0[j], neg_lwp, q);
    float fu = ((j % 6) >= 3) ? u0[j] * gu : gu;
    float fv = ((j % 6) >= 3) ? v0[j] * gv : gv;
    res[j] = (fu >= 0.0f) ? u0[j] : ((fv <= 0.0f) ? v0[j] : m[j]);
  }

  // Candidate selection per problem: argmin with first-tie (strict <)
  // semantics; NaN-first also matches np.argmin (e < NaN is false).
#pragma unroll
  for (int p = 0; p < 2; ++p) {
    const int o = p * 6;
    const float W = wh[p];
    const bool small_wh = (W <= 5.6568542494923806f);   // 4*sqrt(2)
    float cands[7] = { w0[p], W, res[o+1], res[o+2], res[o+3], res[o+4], res[o+5] };
    float beste = eps_fast(cands[0], neg_lwp, W);
    float wB    = cands[0];
#pragma unroll
    for (int k = 1; k < 7; ++k) {
      bool inval = (k == 4) ? (!small_wh) : ((k >= 5) ? small_wh : false);
      float e = inval ? 1e30f : eps_fast(cands[k], neg_lwp, W);
      bool better = (e < beste);
      beste = better ? e : beste;
      wB    = better ? cands[k] : wB;
    }
    const bool branchA = (fmaxf(w0[p], W) < wp);
    const bool branchB = (fmaxf(w0[p], wp) < W);
    float r = branchA ? res[o+0] : (branchB ? wB : w0[p]);
    if (p == 0) *outA = r; else *outB = r;
  }
}

__global__ __launch_bounds__(256)
void cabb_68092411511620_kernel(const float* __restrict__ pred,
                                const float* __restrict__ target,
                                const float* __restrict__ crop,
                                const float* __restrict__ prop,
                                const unsigned char* __restrict__ cases,
                                float* __restrict__ out, int n) {
  int i = blockIdx.x * blockDim.x + threadIdx.x;
  if (i >= 2 * n) return;
  int row  = i >> 1;                // two threads per box: axis 0 and axis 1
  int axis = i & 1;

  const float*         pr = pred   + (size_t)row * 4;
  const float*         tg = target + (size_t)row * 4;
  const float*         pp = prop   + (size_t)row * 4;
  const float*         cr = crop   + (size_t)row * 4;
  const unsigned char* cs = cases  + (size_t)row * 4;

  // gfx1250 global_prefetch_b8: warm the 5 row streams ahead of the
  // dependent scalar loads (free).
  __builtin_prefetch(pr, 0, 0);
  __builtin_prefetch(tg, 0, 0);
  __builtin_prefetch(pp, 0, 0);
  __builtin_prefetch(cr, 0, 0);
  __builtin_prefetch(cs, 0, 0);

  // d_id = axis, o_id = axis+2 ; pred/target cols 2:4 are exp()'d upfront
  float pd  = pr[axis];
  float po  = __expf(pr[axis + 2]);
  float td  = tg[axis];
  float to  = __expf(tg[axis + 2]);
  float p_d = pp[axis];
  float p_o = pp[axis + 2];
  float cra = cr[axis];             // crop[:,axis] == crop[:,d_id]
  bool  lt  = cs[axis * 2 + 0] != 0;
  bool  rb  = cs[axis * 2 + 1] != 0;

  float ca = 0.5f * (p_d + p_o);
  float da = p_o - p_d;
  float inv_da = 1.0f / da;         // one divide shared by b1r / a1l / b2

  // right-crop + left-crop branches share wp = po -> one fused 12-chain loop
  float a1r = td - 0.5f * to;
  float b1r = (cra - ca) * inv_da;
  float a1l = -ca * inv_da;
  float b1l = td + 0.5f * to;
  float wr, wl;
  solve_o1_pair(po,
                2.0f * (pd - a1r), a1r, b1r,    // right
                2.0f * (b1l - pd), a1l, b1l,    // left
                &wr, &wl);
  float dr = a1r + 0.5f * wr;
  float dl = b1l - 0.5f * wl;

  // both-cropped branch: solve_o2(dp=pd, wp=pd, a2=a1l, b2) -> second pair
  float b2  = (cra - ca) * inv_da;
  float wh1 = 2.0f * (pd - a1l);
  float wh2 = 2.0f * (b2 - pd);
  float w1, w2;
  solve_o1_pair(pd, wh1, a1l, b2, wh2, a1l, b2, &w1, &w2);
  bool  trivial = (pd >= fmaxf(wh1, wh2));
  float neg_lpd = -safe_log(pd);
  bool  pick1 = (eps_fast(w1, neg_lpd, wh1) <= eps_fast(w2, neg_lpd, wh2));
  float db = pick1 ? (a1l + 0.5f * w1) : (b2 + 0.5f * w2);
  float wb = pick1 ? w1 : w2;
  db = trivial ? pd : db;
  wb = trivial ? pd : wb;

  bool none_c = (!lt && !rb);
  float d_lab = none_c ? td : (!lt ? dr : (!rb ? dl : db));
  float w_lab = none_c ? to : (!lt ? wr : (!rb ? wl : wb));

  // output = stack([d0, d1, w0, w1], axis=-1)
  out[(size_t)row * 4 + axis]     = d_lab;
  out[(size_t)row * 4 + 2 + axis] = w_lab;
}

extern "C" void kernel_launch(void* const* d_in, const int* in_sizes, int n_in,
                              void* d_out, int out_size, void* d_ws, size_t ws_size,
                              hipStream_t stream) {
  (void)n_in; (void)out_size; (void)d_ws; (void)ws_size;
  // setup_inputs order: img, pred, target, crop_shapes, proposal_list, cases,
  // sampling_results. img + sampling_results are unused by the reference.
  const float*         pred   = (const float*)d_in[1];
  const float*         target = (const float*)d_in[2];
  const float*         crop   = (const float*)d_in[3];
  const float*         prop   = (const float*)d_in[4];
  const unsigned char* cases  = (const unsigned char*)d_in[5];
  float* out = (float*)d_out;

  int n = in_sizes[1] / 4;          // N rows
  int total = 2 * n;                // one thread per (row, axis)
  int block = 256;                  // 8 wave32s per workgroup
  int grid = (total + block - 1) / block;
  cabb_68092411511620_kernel<<<grid, block, 0, stream>>>(pred, target, crop,
                                                         prop, cases, out, n);
}